// DeformParams_16681652978080
// MI455X (gfx1250) — compile-verified
//
#include <hip/hip_runtime.h>
#include <hip/hip_bf16.h>
#include <stdint.h>

// CDNA5 WMMA operand types: f32 16x16x4 -> A,B are 2 VGPRs (v2f), C/D are 8 VGPRs (v8f)
typedef __attribute__((ext_vector_type(2))) float v2f;
typedef __attribute__((ext_vector_type(8))) float v8f;

// ---------------------------------------------------------------------------
// Bulk copy verts -> updated region of d_out. float4 (GLOBAL_LOAD/STORE_B128)
// to saturate the 23.3 TB/s HBM path.
// ---------------------------------------------------------------------------
__global__ __launch_bounds__(256) void copy_b128_kernel(const float4* __restrict__ src,
                                                        float4* __restrict__ dst, int n4) {
    int i = blockIdx.x * blockDim.x + threadIdx.x;
    if (i < n4) dst[i] = src[i];
}

__global__ __launch_bounds__(256) void copy_b32_kernel(const float* __restrict__ src,
                                                       float* __restrict__ dst, int n) {
    int i = blockIdx.x * blockDim.x + threadIdx.x;
    if (i < n) dst[i] = src[i];
}

// ---------------------------------------------------------------------------
// One sample per wave32.
//   A (16x4 f32)  = R_s zero-padded          (rows 0..2 = R rows, K 0..2 = R cols)
//   B (4x16 f32)  = (v_s - v_n[d]) per col d (rows 0..2 = xyz of diff)
//   C = A*B (16x16 f32): row i, col d = rot[i][d]   (only rows 0..2 nonzero)
// Then new_verts[s] = sum_d w[d] * (rot[:,d] + vn[d] + t_s)  via wave reduction.
//
// VGPR layouts per CDNA5 ISA 7.12.2 (wave32):
//   A: lanes 0-15 hold row M=lane; VGPR0=K0,VGPR1=K1. lanes 16-31: VGPR0=K2,VGPR1=K3.
//   B: VGPR0: lanes0-15 -> K=0 (N=lane), lanes16-31 -> K=2; VGPR1: K=1 / K=3.
//   C: VGPR r: lanes0-15 -> row r (N=lane), lanes16-31 -> row r+8.
// ---------------------------------------------------------------------------
__global__ __launch_bounds__(256) void deform_wmma_kernel(
    const float* __restrict__ verts,        // [N,3]
    const float* __restrict__ R,            // [S,3,3]
    const float* __restrict__ t,            // [S,3]
    const float* __restrict__ w,            // [S,16]
    const int*   __restrict__ sampled_ids,  // [S]
    const int*   __restrict__ neighbor_ids, // [S,16]
    float* __restrict__ new_verts,          // [S,3]   (d_out part 0)
    float* __restrict__ updated,            // [N,3]   (d_out part 1)
    int S)
{
    const int lane = threadIdx.x & 31;
    const int wave = blockIdx.x * (blockDim.x >> 5) + (threadIdx.x >> 5);
    if (wave >= S) return;                  // wave-uniform: EXEC stays all-1s
    const int s  = wave;
    const int m  = lane & 15;               // neighbor index d (both half-waves)
    const bool hi = lane >= 16;

    const int sid = sampled_ids[s];
    const int nid = neighbor_ids[s * 16 + m];

    // Gathers hit L2 (verts = 24 MB << 192 MB L2)
    const float vn0 = verts[(size_t)nid * 3 + 0];
    const float vn1 = verts[(size_t)nid * 3 + 1];
    const float vn2 = verts[(size_t)nid * 3 + 2];
    const float v0  = verts[(size_t)sid * 3 + 0];
    const float v1  = verts[(size_t)sid * 3 + 1];
    const float v2  = verts[(size_t)sid * 3 + 2];

    const float dx = v0 - vn0, dy = v1 - vn1, dz = v2 - vn2;

    // Build A = R padded to 16x4 (selects lower to v_cndmask, no EXEC change)
    const float* Rs = R + (size_t)s * 9;
    const int row  = (m < 3) ? m : 0;
    const float rk0 = Rs[row * 3 + (hi ? 2 : 0)];        // K = 0 or 2 (< 3, valid)
    const float rk1 = hi ? 0.0f : Rs[row * 3 + 1];       // K = 1 valid, K = 3 -> 0
    v2f a; a.x = (m < 3) ? rk0 : 0.0f;
           a.y = (m < 3) ? rk1 : 0.0f;

    // Build B = diffs padded to 4x16 (column d = this lane's neighbor)
    v2f b; b.x = hi ? dz : dx;                           // K=2 : K=0
           b.y = hi ? 0.0f : dy;                         // K=3(0) : K=1

    v8f c = {};
    // D = A x B + 0 : emits v_wmma_f32_16x16x4_f32
    c = __builtin_amdgcn_wmma_f32_16x16x4_f32(
            /*neg_a=*/false, a, /*neg_b=*/false, b,
            /*c_mod=*/(short)0, c, /*reuse_a=*/false, /*reuse_b=*/false);

    // Lanes 0-15: c[0..2] = rot[0..2] for neighbor d = lane.
    const float wt = w[(size_t)s * 16 + m];
    const float t0 = t[(size_t)s * 3 + 0];
    const float t1 = t[(size_t)s * 3 + 1];
    const float t2 = t[(size_t)s * 3 + 2];

    float acc0 = hi ? 0.0f : wt * (c[0] + vn0 + t0);
    float acc1 = hi ? 0.0f : wt * (c[1] + vn1 + t1);
    float acc2 = hi ? 0.0f : wt * (c[2] + vn2 + t2);

    // Full-wave butterfly reduction (hi lanes contribute 0)
    #pragma unroll
    for (int off = 16; off >= 1; off >>= 1) {
        acc0 += __shfl_xor(acc0, off, 32);
        acc1 += __shfl_xor(acc1, off, 32);
        acc2 += __shfl_xor(acc2, off, 32);
    }

    if (lane == 0) {
        new_verts[(size_t)s * 3 + 0] = acc0;
        new_verts[(size_t)s * 3 + 1] = acc1;
        new_verts[(size_t)s * 3 + 2] = acc2;
        updated[(size_t)sid * 3 + 0] = acc0;   // scatter: updated[sampled_ids] = new
        updated[(size_t)sid * 3 + 1] = acc1;
        updated[(size_t)sid * 3 + 2] = acc2;
    }
}

extern "C" void kernel_launch(void* const* d_in, const int* in_sizes, int n_in,
                              void* d_out, int out_size, void* d_ws, size_t ws_size,
                              hipStream_t stream) {
    const float* verts        = (const float*)d_in[0];   // [N,3]
    const float* R            = (const float*)d_in[1];   // [S,3,3]
    const float* t            = (const float*)d_in[2];   // [S,3]
    const float* w            = (const float*)d_in[3];   // [S,16]
    const int*   sampled_ids  = (const int*)d_in[4];     // [S]
    const int*   neighbor_ids = (const int*)d_in[5];     // [S,16]

    const int S   = in_sizes[4];       // 500,000
    const int nv3 = in_sizes[0];       // N_VERTS * 3 = 6,000,000

    float* new_verts = (float*)d_out;                       // [S,3]
    float* updated   = (float*)d_out + (size_t)S * 3;       // [N,3]

    // 1) Bulk copy verts -> updated (vectorized if 16B-aligned, else scalar).
    const bool vec_ok =
        ((reinterpret_cast<uintptr_t>(verts)   & 15u) == 0) &&
        ((reinterpret_cast<uintptr_t>(updated) & 15u) == 0) &&
        ((nv3 & 3) == 0);
    if (vec_ok) {
        const int n4 = nv3 >> 2;
        copy_b128_kernel<<<(n4 + 255) / 256, 256, 0, stream>>>(
            (const float4*)verts, (float4*)updated, n4);
    } else {
        copy_b32_kernel<<<(nv3 + 255) / 256, 256, 0, stream>>>(verts, updated, nv3);
    }

    // 2) Deform: one sample per wave32, 8 waves per 256-thread block.
    const int blocks = (S + 7) / 8;
    deform_wmma_kernel<<<blocks, 256, 0, stream>>>(
        verts, R, t, w, sampled_ids, neighbor_ids, new_verts, updated, S);
}